// GraphSAGEEncoder_65910568124791
// MI455X (gfx1250) — compile-verified
//
#include <hip/hip_runtime.h>

// GraphSAGE encoder for MI455X (gfx1250, wave32).
// GEMMs use V_WMMA_F32_16X16X4_F32 (full f32 precision, matrix pipe).
// Edge aggregation uses global_atomic_add_f32 into an L2-resident buffer.
// BatchNorm column stats fused into GEMM epilogues via f32 atomics.

typedef __attribute__((ext_vector_type(2))) float v2f;
typedef __attribute__((ext_vector_type(8))) float v8f;

#define WMMA_F32(a, b, c) \
  __builtin_amdgcn_wmma_f32_16x16x4_f32(false, (a), false, (b), (short)0, (c), false, false)

// ---------------------------------------------------------------- zero
__global__ __launch_bounds__(256) void k_zero(float* __restrict__ p, int n4) {
  int i = blockIdx.x * 256 + threadIdx.x;
  if (i < n4) ((float4*)p)[i] = make_float4(0.f, 0.f, 0.f, 0.f);
}

// ------------------------------------------------- edge gather/scatter
// one wave per edge; lane handles 4 consecutive features (float4 gather,
// 4x global_atomic_add_f32 scatter). lane0 optionally counts in-degree.
__global__ __launch_bounds__(256) void k_edge_agg(
    const float* __restrict__ feat, const int* __restrict__ src,
    const int* __restrict__ dst, float* __restrict__ agg,
    float* __restrict__ cnt, int E, int add_cnt) {
  int wave = (blockIdx.x * 256 + threadIdx.x) >> 5;
  int lane = threadIdx.x & 31;
  if (wave >= E) return;
  int s = src[wave];
  int d = dst[wave];
  const float4 v = *(const float4*)(feat + (size_t)s * 128 + lane * 4);
  float* ap = agg + (size_t)d * 128 + lane * 4;
  atomicAdd(ap + 0, v.x);
  atomicAdd(ap + 1, v.y);
  atomicAdd(ap + 2, v.z);
  atomicAdd(ap + 3, v.w);
  if (add_cnt && lane == 0) atomicAdd(cnt + d, 1.0f);
}

// ------------------------------------------------- SAGE layer-1 GEMM
// P = X@Wself + (AGG/max(cnt,1))@Wneigh + bias ; accumulate col sum/sumsq.
// Block = 16 rows x 128 cols; 8 waves, one 16x16 tile each; K-loop step 4.
__global__ __launch_bounds__(256) void k_sage_gemm1(
    const float* __restrict__ X, const float* __restrict__ AGG,
    const float* __restrict__ CNT, const float* __restrict__ Wself,
    const float* __restrict__ Wneigh, const float* __restrict__ bias,
    float* __restrict__ P, float* __restrict__ colsum, float* __restrict__ colsq) {
  __shared__ float ldsX[16 * 132];
  __shared__ float ldsM[16 * 132];
  const int r0 = blockIdx.x * 16;
  for (int j = threadIdx.x; j < 512; j += 256) {  // 16 rows x 32 float4
    int row = j >> 5;
    int c4 = (j & 31) << 2;
    size_t g = (size_t)(r0 + row) * 128 + c4;
    float4 xv = *(const float4*)(X + g);
    float4 av = *(const float4*)(AGG + g);
    float inv = 1.0f / fmaxf(CNT[r0 + row], 1.0f);
    av.x *= inv; av.y *= inv; av.z *= inv; av.w *= inv;
    int l = row * 132 + c4;
    *(float4*)(ldsX + l) = xv;
    *(float4*)(ldsM + l) = av;
  }
  __syncthreads();
  const int lane = threadIdx.x & 31;
  const int half = lane >> 4;
  const int m = lane & 15;
  const int c0 = (threadIdx.x >> 5) * 16;
  const int col = c0 + m;
  v8f acc = {};
  for (int kk = 0; kk < 128; kk += 4) {
    const int ka = kk + 2 * half;
    v2f aX = {ldsX[m * 132 + ka], ldsX[m * 132 + ka + 1]};
    v2f aM = {ldsM[m * 132 + ka], ldsM[m * 132 + ka + 1]};
    v2f bS = {Wself[(size_t)ka * 128 + col], Wself[(size_t)(ka + 1) * 128 + col]};
    v2f bN = {Wneigh[(size_t)ka * 128 + col], Wneigh[(size_t)(ka + 1) * 128 + col]};
    acc = WMMA_F32(aX, bS, acc);
    acc = WMMA_F32(aM, bN, acc);
  }
  const float bc = bias[col];
  float s = 0.f, sq = 0.f;
#pragma unroll
  for (int i = 0; i < 8; ++i) {
    float v = acc[i] + bc;
    P[(size_t)(r0 + i + 8 * half) * 128 + col] = v;
    s += v;
    sq += v * v;
  }
  atomicAdd(colsum + col, s);
  atomicAdd(colsq + col, sq);
}

// ------------------------------------------------- SAGE layer-2 GEMM
// P2 = H@Wself2 + mean2@Wneigh2 + b2 ; S = H@Wskip + bskip ; stats for both.
__global__ __launch_bounds__(256) void k_sage_gemm2(
    const float* __restrict__ H, const float* __restrict__ AGG,
    const float* __restrict__ CNT, const float* __restrict__ Wself,
    const float* __restrict__ Wneigh, const float* __restrict__ b2,
    const float* __restrict__ Wskip, const float* __restrict__ bskip,
    float* __restrict__ P2, float* __restrict__ S2,
    float* __restrict__ colsumP, float* __restrict__ colsqP,
    float* __restrict__ colsumS, float* __restrict__ colsqS) {
  __shared__ float ldsH[16 * 132];
  __shared__ float ldsM[16 * 132];
  const int r0 = blockIdx.x * 16;
  for (int j = threadIdx.x; j < 512; j += 256) {
    int row = j >> 5;
    int c4 = (j & 31) << 2;
    size_t g = (size_t)(r0 + row) * 128 + c4;
    float4 hv = *(const float4*)(H + g);
    float4 av = *(const float4*)(AGG + g);
    float inv = 1.0f / fmaxf(CNT[r0 + row], 1.0f);
    av.x *= inv; av.y *= inv; av.z *= inv; av.w *= inv;
    int l = row * 132 + c4;
    *(float4*)(ldsH + l) = hv;
    *(float4*)(ldsM + l) = av;
  }
  __syncthreads();
  const int lane = threadIdx.x & 31;
  const int half = lane >> 4;
  const int m = lane & 15;
  const int c0 = (threadIdx.x >> 5) * 16;
  const int col = c0 + m;
  v8f accP = {};
  v8f accS = {};
  for (int kk = 0; kk < 128; kk += 4) {
    const int ka = kk + 2 * half;
    v2f aH = {ldsH[m * 132 + ka], ldsH[m * 132 + ka + 1]};
    v2f aM = {ldsM[m * 132 + ka], ldsM[m * 132 + ka + 1]};
    v2f bS = {Wself[(size_t)ka * 128 + col], Wself[(size_t)(ka + 1) * 128 + col]};
    v2f bN = {Wneigh[(size_t)ka * 128 + col], Wneigh[(size_t)(ka + 1) * 128 + col]};
    v2f bK = {Wskip[(size_t)ka * 128 + col], Wskip[(size_t)(ka + 1) * 128 + col]};
    accP = WMMA_F32(aH, bS, accP);
    accP = WMMA_F32(aM, bN, accP);
    accS = WMMA_F32(aH, bK, accS);
  }
  const float bcP = b2[col];
  const float bcS = bskip[col];
  float sP = 0.f, qP = 0.f, sS = 0.f, qS = 0.f;
#pragma unroll
  for (int i = 0; i < 8; ++i) {
    size_t off = (size_t)(r0 + i + 8 * half) * 128 + col;
    float vP = accP[i] + bcP;
    float vS = accS[i] + bcS;
    P2[off] = vP;
    S2[off] = vS;
    sP += vP; qP += vP * vP;
    sS += vS; qS += vS * vS;
  }
  atomicAdd(colsumP + col, sP);
  atomicAdd(colsqP + col, qP);
  atomicAdd(colsumS + col, sS);
  atomicAdd(colsqS + col, qS);
}

// ------------------------------------------- BN finalize: fold to scale/shift
__global__ void k_bn_finalize(const float* __restrict__ sum,
                              const float* __restrict__ sq,
                              const float* __restrict__ g,
                              const float* __restrict__ be,
                              float* __restrict__ a, float* __restrict__ sh,
                              float invN) {
  int c = threadIdx.x;  // 128 threads
  float m = sum[c] * invN;
  float v = sq[c] * invN - m * m;
  float is = rsqrtf(v + 1e-5f);
  float aa = g[c] * is;
  a[c] = aa;
  sh[c] = be[c] - m * aa;
}

// ------------------------------------------- elementwise BN+ReLU in place
__global__ __launch_bounds__(256) void k_bn_relu_apply(
    float* __restrict__ H, const float* __restrict__ a,
    const float* __restrict__ sh, int n4) {
  int i = blockIdx.x * 256 + threadIdx.x;
  if (i >= n4) return;
  float4 v = ((float4*)H)[i];
  int c = (i << 2) & 127;
  v.x = fmaxf(fmaf(a[c + 0], v.x, sh[c + 0]), 0.f);
  v.y = fmaxf(fmaf(a[c + 1], v.y, sh[c + 1]), 0.f);
  v.z = fmaxf(fmaf(a[c + 2], v.z, sh[c + 2]), 0.f);
  v.w = fmaxf(fmaf(a[c + 3], v.w, sh[c + 3]), 0.f);
  ((float4*)H)[i] = v;
}

// ------------------------------------------- out = relu(bn2(P2) + bn3(S))
__global__ __launch_bounds__(256) void k_final(
    const float* __restrict__ P2, const float* __restrict__ S,
    const float* __restrict__ a2, const float* __restrict__ s2,
    const float* __restrict__ a3, const float* __restrict__ s3,
    float* __restrict__ out, int n4) {
  int i = blockIdx.x * 256 + threadIdx.x;
  if (i >= n4) return;
  float4 p = ((const float4*)P2)[i];
  float4 s = ((const float4*)S)[i];
  int c = (i << 2) & 127;
  float4 o;
  o.x = fmaxf(fmaf(a2[c + 0], p.x, s2[c + 0]) + fmaf(a3[c + 0], s.x, s3[c + 0]), 0.f);
  o.y = fmaxf(fmaf(a2[c + 1], p.y, s2[c + 1]) + fmaf(a3[c + 1], s.y, s3[c + 1]), 0.f);
  o.z = fmaxf(fmaf(a2[c + 2], p.z, s2[c + 2]) + fmaf(a3[c + 2], s.z, s3[c + 2]), 0.f);
  o.w = fmaxf(fmaf(a2[c + 3], p.w, s2[c + 3]) + fmaf(a3[c + 3], s.w, s3[c + 3]), 0.f);
  ((float4*)out)[i] = o;
}

// ----------------------------------------------------------------------
extern "C" void kernel_launch(void* const* d_in, const int* in_sizes, int n_in,
                              void* d_out, int out_size, void* d_ws, size_t ws_size,
                              hipStream_t stream) {
  const float* x       = (const float*)d_in[0];
  const int*   src     = (const int*)d_in[1];
  const int*   dst     = (const int*)d_in[2];
  const float* w_self1 = (const float*)d_in[3];
  const float* w_neigh1= (const float*)d_in[4];
  const float* b1      = (const float*)d_in[5];
  const float* w_self2 = (const float*)d_in[6];
  const float* w_neigh2= (const float*)d_in[7];
  const float* b2      = (const float*)d_in[8];
  const float* w_skip  = (const float*)d_in[9];
  const float* b_skip  = (const float*)d_in[10];
  const float* g1      = (const float*)d_in[11];
  const float* be1     = (const float*)d_in[12];
  const float* g2      = (const float*)d_in[13];
  const float* be2     = (const float*)d_in[14];
  const float* g3      = (const float*)d_in[15];
  const float* be3     = (const float*)d_in[16];
  float* out = (float*)d_out;

  const int N = in_sizes[0] / 128;  // 100000
  const int E = in_sizes[1];        // 1600000
  const size_t ND = (size_t)N * 128;

  // workspace layout (floats)
  float* AGG = (float*)d_ws;       // N*128 (reused by both layers)
  float* CNT = AGG + ND;           // N
  float* H1  = CNT + N;            // N*128 (pre-BN then post-BN in place)
  float* P2  = H1 + ND;            // N*128
  float* S   = P2 + ND;            // N*128
  float* ST  = S + ND;             // 12*128 stats/params
  float* csum1 = ST;        float* csq1 = ST + 128;
  float* csum2 = ST + 256;  float* csq2 = ST + 384;
  float* csumS = ST + 512;  float* csqS = ST + 640;
  float* a1 = ST + 768;     float* s1 = ST + 896;
  float* a2 = ST + 1024;    float* s2 = ST + 1152;
  float* a3 = ST + 1280;    float* s3 = ST + 1408;

  const float invN = 1.0f / (float)N;
  const int nd4 = (int)(ND / 4);
  const int gridEW = (E + 7) / 8;      // one wave32 per edge, 8 waves/block
  const int gridRows = N / 16;         // 16-row tiles (N divisible by 16)
  const int gridElem = (nd4 + 255) / 256;

  // ---- layer 1
  {
    int z4 = (int)((ND + N) / 4);      // AGG + CNT contiguous
    k_zero<<<(z4 + 255) / 256, 256, 0, stream>>>(AGG, z4);
    k_zero<<<2, 256, 0, stream>>>(ST, 1536 / 4);
  }
  k_edge_agg<<<gridEW, 256, 0, stream>>>(x, src, dst, AGG, CNT, E, 1);
  k_sage_gemm1<<<gridRows, 256, 0, stream>>>(x, AGG, CNT, w_self1, w_neigh1, b1,
                                             H1, csum1, csq1);
  k_bn_finalize<<<1, 128, 0, stream>>>(csum1, csq1, g1, be1, a1, s1, invN);
  k_bn_relu_apply<<<gridElem, 256, 0, stream>>>(H1, a1, s1, nd4);

  // ---- layer 2 + skip
  k_zero<<<(nd4 + 255) / 256, 256, 0, stream>>>(AGG, nd4);
  k_edge_agg<<<gridEW, 256, 0, stream>>>(H1, src, dst, AGG, CNT, E, 0);
  k_sage_gemm2<<<gridRows, 256, 0, stream>>>(H1, AGG, CNT, w_self2, w_neigh2, b2,
                                             w_skip, b_skip, P2, S,
                                             csum2, csq2, csumS, csqS);
  k_bn_finalize<<<1, 128, 0, stream>>>(csum2, csq2, g2, be2, a2, s2, invN);
  k_bn_finalize<<<1, 128, 0, stream>>>(csumS, csqS, g3, be3, a3, s3, invN);
  k_final<<<gridElem, 256, 0, stream>>>(P2, S, a2, s2, a3, s3, out, nd4);
}